// DetLoss_3762391351632
// MI455X (gfx1250) — compile-verified
//
#include <hip/hip_runtime.h>
#include <cstdint>

#define NUM_C 40
#define NUM_M 32

__device__ __forceinline__ float iou_fn(float ax1, float ay1, float ax2, float ay2, float aarea,
                                        float b0, float b1, float b2, float b3, float barea) {
    float iw = fminf(ax2, b2) - fmaxf(ax1, b0); iw = fmaxf(iw, 0.0f);
    float ih = fminf(ay2, b3) - fmaxf(ay1, b1); ih = fmaxf(ih, 0.0f);
    float inter = iw * ih;
    float ua = fmaxf(aarea + barea - inter, 1e-8f);
    // inter == 0 for the vast majority of pairs -> exact 0 regardless of rcp rounding.
    return inter * __builtin_amdgcn_rcpf(ua);
}

__device__ __forceinline__ double shfl_down_d(double x, int off) {
    union { double d; int i[2]; } u; u.d = x;
    u.i[0] = __shfl_down(u.i[0], off, 32);
    u.i[1] = __shfl_down(u.i[1], off, 32);
    return u.d;
}

// CDNA5 async global->LDS copy (16B per active lane), tracked by ASYNCcnt.
__device__ __forceinline__ void async_ld_b128(unsigned lds_byte_off, const void* gptr) {
    asm volatile("global_load_async_to_lds_b128 %0, %1, off"
                 :: "v"(lds_byte_off),
                    "v"((unsigned long long)(uintptr_t)gptr)
                 : "memory");
}
__device__ __forceinline__ void wait_async0() {
    asm volatile("s_wait_asynccnt 0" ::: "memory");
}

// ---------------- prep: annotation -> bbox/area/valid table; zero accumulators ----------------
__global__ void prep_kernel(const float* __restrict__ ann, float* __restrict__ sann,
                            double* __restrict__ accd, int* __restrict__ accn, int B) {
    int t = threadIdx.x;                       // launched with >= B*NUM_M threads
    if (t < 2 * B) accd[t] = 0.0;
    if (t < B)     accn[t] = 0;
    if (t < B * NUM_M) {
        const float* r = ann + (size_t)t * 5;
        float cx = r[0], cy = r[1], angv = r[2], ln = r[3], clsv = r[4];
        float valid = (clsv != -1.0f) ? 1.0f : 0.0f;
        float dx = fabsf(0.5f * ln * cosf(angv));
        float dy = fabsf(0.5f * ln * sinf(angv));
        float b0 = cx - dx, b1 = cy - dy, b2 = cx + dx, b3 = cy + dy;
        float area = (b2 - b0) * (b3 - b1);
        float* o = sann + (size_t)t * 12;
        o[0] = b0; o[1] = b1; o[2] = b2; o[3] = b3; o[4] = area; o[5] = valid;
        o[6] = cx; o[7] = cy; o[8] = angv; o[9] = ln; o[10] = clsv; o[11] = 0.0f;
    }
}

// ---------------- col: per (image, annotation) column max/argmax over all anchors -------------
__global__ __launch_bounds__(256) void col_kernel(const float* __restrict__ anchors,
                                                  const float* __restrict__ sann,
                                                  int* __restrict__ ovA, int A) {
    __shared__ float sv[256];
    __shared__ int   si[256];
    int bj = blockIdx.x;                       // b*NUM_M + j
    const float* an = sann + (size_t)bj * 12;
    int t = threadIdx.x;
    if (an[5] == 0.0f) {                       // invalid annotation: never an override
        if (t == 0) ovA[bj] = -1;
        return;
    }
    float b0 = an[0], b1 = an[1], b2 = an[2], b3 = an[3], area = an[4];
    float best = -1.0f; int bi = 0;
    for (int a = t; a < A; a += 256) {
        float4 v = ((const float4*)anchors)[a];
        float aarea = (v.z - v.x) * (v.w - v.y);
        float io = iou_fn(v.x, v.y, v.z, v.w, aarea, b0, b1, b2, b3, area);
        if (io > best) { best = io; bi = a; }  // strict > keeps first index within thread
    }
    sv[t] = best; si[t] = bi;
    __syncthreads();
    for (int s = 128; s > 0; s >>= 1) {
        if (t < s) {
            float ov = sv[t + s]; int oi = si[t + s];
            if (ov > sv[t] || (ov == sv[t] && oi < si[t])) { sv[t] = ov; si[t] = oi; }
        }
        __syncthreads();
    }
    if (t == 0) ovA[bj] = (sv[0] < 0.5f) ? si[0] : -1;   // "do" condition of the scan
}

// ---------------- main: stream classifications via async LDS; focal + regression --------------
__global__ __launch_bounds__(256) void main_kernel(const float* __restrict__ cls,
                                                   const float* __restrict__ regr,
                                                   const float* __restrict__ anchors,
                                                   const float* __restrict__ sann,
                                                   const int* __restrict__ ovA,
                                                   double* __restrict__ accd,
                                                   int* __restrict__ accn,
                                                   int A, int B) {
    __shared__ float4 stage[256 * (NUM_C / 4)];   // 40 KB staging tile (lane-private slices)
    __shared__ float  sA[NUM_M * 12];
    __shared__ int    sOv[NUM_M];
    __shared__ double redC[8], redR[8];
    __shared__ int    redN[8];

    const int t = threadIdx.x;
    const int b = blockIdx.y;
    const int a = blockIdx.x * 256 + t;
    const bool active = (a < A);
    const int  aa = active ? a : (A - 1);

    // Kick off async staging of this anchor's 40-class row (10 x b128 per lane).
    const float* crow = cls + ((size_t)b * A + aa) * NUM_C;
    unsigned lds0 = (unsigned)(unsigned long long)(uintptr_t)&stage[0] + (unsigned)t * (NUM_C * 4u);
#pragma unroll
    for (int i = 0; i < NUM_C / 4; ++i)
        async_ld_b128(lds0 + 16u * i, crow + 4 * i);

    // Preload annotation table + override table (overlaps with in-flight async DMA).
    for (int i = t; i < NUM_M * 12; i += 256) sA[i] = sann[(size_t)b * NUM_M * 12 + i];
    if (t < NUM_M) sOv[t] = ovA[b * NUM_M + t];
    __syncthreads();

    // Row IoU max / first-argmax over the 32 annotations (overlaps the DMA too).
    float4 av = ((const float4*)anchors)[aa];
    float aarea = (av.z - av.x) * (av.w - av.y);
    float best = -2.0f; int bj = 0;
#pragma unroll
    for (int j = 0; j < NUM_M; ++j) {
        const float* an = &sA[j * 12];
        float e = (an[5] != 0.0f)
                      ? iou_fn(av.x, av.y, av.z, av.w, aarea, an[0], an[1], an[2], an[3], an[4])
                      : -1.0f;
        if (e > best) { best = e; bj = j; }
    }

    // Sequential-scan override: highest j targeting this anchor wins.
    int ovj = -1;
#pragma unroll
    for (int j = 0; j < NUM_M; ++j)
        if (sOv[j] == a) ovj = j;              // inactive lanes: a >= A never matches

    bool pos = false; int annj = 0;
    if (ovj >= 0)            { pos = true; annj = ovj; }
    else if (best >= 0.5f)   { pos = true; annj = bj; }
    const bool neg = (!pos) && (best < 0.4f);
    const bool ign = (!pos) && (!neg);
    const int  kcls = pos ? (int)sA[annj * 12 + 10] : -1;

    // Focal loss over 40 classes from the async-staged LDS tile.
    wait_async0();
    float fsum = 0.0f;
    if (pos && active) {
        // Rare path: one class is the target, exact per-class select.
#pragma unroll
        for (int i = 0; i < NUM_C / 4; ++i) {
            float4 v = stage[t * (NUM_C / 4) + i];
            float vv[4] = { v.x, v.y, v.z, v.w };
#pragma unroll
            for (int e = 0; e < 4; ++e) {
                float c = fminf(fmaxf(vv[e], 1e-4f), 1.0f - 1e-4f);
                bool ist = ((i * 4 + e) == kcls);
                float u = 1.0f - c;
                float larg = ist ? c : u;
                float sq   = ist ? u : c;
                float af   = ist ? 0.25f : 0.75f;
                fsum += af * (sq * sq * (-__logf(larg)));
            }
        }
    } else {
        // Hot path (negatives / ignored / inactive): term = c^2 * (K * log2(1-c)),
        // K = -0.75*ln2, forced to 0 for ignored/inactive lanes (log2 arg is clamped
        // finite, so 0 * finite == 0 -- no per-element select needed).
        const float K = (ign || !active) ? 0.0f : -0.75f * 0.69314718055994531f;
#pragma unroll
        for (int i = 0; i < NUM_C / 4; ++i) {
            float4 v = stage[t * (NUM_C / 4) + i];
            float vv[4] = { v.x, v.y, v.z, v.w };
#pragma unroll
            for (int e = 0; e < 4; ++e) {
                float c = fminf(fmaxf(vv[e], 1e-4f), 1.0f - 1e-4f);
                float u = 1.0f - c;
                fsum += (c * c) * (K * __log2f(u));   // v_log_f32 co-issues with VALU
            }
        }
    }
    double cacc = (double)fsum;

    // Regression (positive anchors only).
    double racc = 0.0; int np = 0;
    if (pos && active) {
        const float* an = &sA[annj * 12];
        float cx = an[6], cy = an[7], angv = an[8], ln = an[9];
        float acx = (av.x + av.z) * 0.5f, acy = (av.y + av.w) * 0.5f;
        float aw = av.z - av.x, ah = av.w - av.y;
        float alen = sqrtf(aw * aw + ah * ah);
        float4 r = ((const float4*)regr)[(size_t)b * A + aa];
        float tdx = ((cx - acx) / aw) / 0.5f;
        float tdy = ((cy - acy) / ah) / 0.5f;
        float gl  = fmaxf(ln, 1.0f);
        float tdl = __logf(gl / alen) / 0.5f;
        float d0 = fabsf(tdx - r.x);
        float d1 = fabsf(tdy - r.y);
        float d2 = fabsf(__sinf(angv - r.z));
        float d3 = fabsf(tdl - r.w);
        auto sl = [](float d) { return (d <= 1.0f) ? 0.5f * d * d : d - 0.5f; };
        float per = 5.0f * (sl(d0) + sl(d1) + sl(d2) + sl(d3));
        racc = (double)per;
        np = 1;
    }

    // Wave32 shuffle reduction, then block LDS reduction, then 3 atomics per block.
#pragma unroll
    for (int off = 16; off > 0; off >>= 1) {
        cacc += shfl_down_d(cacc, off);
        racc += shfl_down_d(racc, off);
        np   += __shfl_down(np, off, 32);
    }
    int w = t >> 5;
    if ((t & 31) == 0) { redC[w] = cacc; redR[w] = racc; redN[w] = np; }
    __syncthreads();
    if (t == 0) {
        double c2 = 0.0, r2 = 0.0; int n2 = 0;
#pragma unroll
        for (int i = 0; i < 8; ++i) { c2 += redC[i]; r2 += redR[i]; n2 += redN[i]; }
        atomicAdd(&accd[b], c2);
        atomicAdd(&accd[B + b], r2);
        atomicAdd(&accn[b], n2);
    }
}

// ---------------- final: combine per-image sums into the scalar loss ----------------
__global__ void final_kernel(const double* __restrict__ accd, const int* __restrict__ accn,
                             float* __restrict__ out, int B) {
    if (threadIdx.x == 0) {
        double cs = 0.0, rs = 0.0;
        for (int b = 0; b < B; ++b) {
            int np = accn[b];
            double d = (np > 1) ? (double)np : 1.0;
            cs += accd[b] / d;
            rs += (np > 0) ? (accd[B + b] / d) : 0.0;
        }
        out[0] = (float)(cs / (double)B + rs / (double)B);
    }
}

extern "C" void kernel_launch(void* const* d_in, const int* in_sizes, int n_in,
                              void* d_out, int out_size, void* d_ws, size_t ws_size,
                              hipStream_t stream) {
    const float* cls     = (const float*)d_in[0];   // (B, A, 40)
    const float* regr    = (const float*)d_in[1];   // (B, A, 4)
    const float* anchors = (const float*)d_in[2];   // (A, 4)
    const float* ann     = (const float*)d_in[3];   // (B, 32, 5)
    float* out = (float*)d_out;

    const int A = in_sizes[2] / 4;
    const int B = in_sizes[3] / (NUM_M * 5);

    char* p = (char*)d_ws;
    float* sann = (float*)p;  p += (size_t)B * NUM_M * 12 * sizeof(float);
    int*   ovA  = (int*)p;    p += (size_t)B * NUM_M * sizeof(int);
    p = (char*)(((uintptr_t)p + 15) & ~(uintptr_t)15);
    double* accd = (double*)p; p += (size_t)2 * B * sizeof(double);
    int*    accn = (int*)p;

    prep_kernel<<<1, 256, 0, stream>>>(ann, sann, accd, accn, B);
    col_kernel<<<B * NUM_M, 256, 0, stream>>>(anchors, sann, ovA, A);
    dim3 grid((A + 255) / 256, B);
    main_kernel<<<grid, 256, 0, stream>>>(cls, regr, anchors, sann, ovA, accd, accn, A, B);
    final_kernel<<<1, 32, 0, stream>>>(accd, accn, out, B);
}